// InverseRealVectorSHT_2731599200381
// MI455X (gfx1250) — compile-verified
//
#include <hip/hip_runtime.h>
#include <math.h>

// ---------------- problem constants ----------------
#define LDIM   360         // LMAX
#define MM     361         // MMAX
#define NLATC  361         // NLAT
#define NCOL   722         // 2*NLAT (stage-1 N)
#define NT1    46          // ceil(722/16)
#define MPAD   364         // MMAX padded to mult of 4
#define XK     728         // 2*MPAD (stage-2 K)
#define ROWS2  46208       // B*C*2*NLAT (stage-2 M), divisible by 32
#define TM2    1444        // ROWS2 / 32
#define NLON_C 720
#define XLSTR  722         // l-stride in x (MM*2 floats)

typedef __attribute__((ext_vector_type(2))) float v2f;
typedef __attribute__((ext_vector_type(8))) float v8f;

// D = A(16x4, f32) * B(4x16, f32) + C(16x16, f32)  ->  v_wmma_f32_16x16x4_f32
__device__ inline v8f wmma4(v2f a, v2f b, v8f c) {
    return __builtin_amdgcn_wmma_f32_16x16x4_f32(
        false, a, false, b, (short)0, c, false, false);
}

// ---------------- stage 1: per-m complex-combined GEMMs ----------------
// For each (m, g): C(32x722) = A(32x720) * B(720x722)
//   g0: A=[x0r|x1i], B=[[d0, d1],[-d1,-d0]] -> [srl|tim]
//   g1: A=[x0i|x1r], B=[[d0,-d1],[ d1,-d0]] -> [sim|trl]
// Signs and column routing are per-lane constants -> applied in the epilogue.
// grid = (6, 361, 2), block = 256 (8 waves); wave -> one 32x16 tile
__global__ void __launch_bounds__(256)
sht_stage1(const float* __restrict__ x, const float* __restrict__ dpct,
           float* __restrict__ X /* pair-interleaved [XK/2][ROWS2][2] */) {
    const int lane = threadIdx.x & 31;
    const int wave = threadIdx.x >> 5;
    const int half = lane >> 4;
    const int r    = lane & 15;
    const int m    = blockIdx.y;
    const int g    = blockIdx.z;
    const int nt   = blockIdx.x * 8 + wave;
    if (nt >= NT1) return;                        // wave-uniform exit
    const int n0 = nt * 16;

    // ---- per-lane column descriptor (loop-invariant) ----
    const int col   = n0 + r;
    const int valid = (col < NCOL);
    const int right = valid && (col >= NLATC);
    const int kc    = valid ? (right ? col - NLATC : col) : 0;

    const int cT = right ? 1 : 0;                 // d-index, top rows (l<360)
    const int cB = right ? 0 : 1;                 // d-index, bottom rows
    float sT, sB;                                 // per-lane column signs
    if (!valid)      { sT = 0.f;                       sB = 0.f; }
    else if (!right) { sT = 1.f;                       sB = (g == 0) ? -1.f : 1.f; }
    else             { sT = (g == 0) ? 1.f : -1.f;     sB = -1.f; }

    // ---- base pointers (fold 2*half into the k offset) ----
    const float* pBT = dpct + (((size_t)cT * MM + m) * LDIM + 2 * half) * NLATC + kc;
    const float* pBB = dpct + (((size_t)cB * MM + m) * LDIM + 2 * half) * NLATC + kc;

    // A[row, lp]: x offset = ((row*2 + j)*LDIM + l)*XLSTR + 2*m + ri
    const size_t abase = (size_t)2 * m;
    const float* pAT0 = x + ((size_t)(2 * (r +  0) + 0) * LDIM + 2 * half) * XLSTR + abase + g;
    const float* pAT1 = x + ((size_t)(2 * (r + 16) + 0) * LDIM + 2 * half) * XLSTR + abase + g;
    const float* pAB0 = x + ((size_t)(2 * (r +  0) + 1) * LDIM + 2 * half) * XLSTR + abase + (g ^ 1);
    const float* pAB1 = x + ((size_t)(2 * (r + 16) + 1) * LDIM + 2 * half) * XLSTR + abase + (g ^ 1);

    v8f accT0 = {}, accT1 = {}, accB0 = {}, accB1 = {};

    // ---- top segment: l in [0,360) ----
    #pragma unroll 2
    for (int i = 0; i < LDIM / 4; ++i) {
        v2f a0, a1, b;
        b.x  = pBT[0];       b.y  = pBT[NLATC];
        a0.x = pAT0[0];      a0.y = pAT0[XLSTR];
        a1.x = pAT1[0];      a1.y = pAT1[XLSTR];
        accT0 = wmma4(a0, b, accT0);
        accT1 = wmma4(a1, b, accT1);
        pBT  += 4 * NLATC;
        pAT0 += 4 * XLSTR;  pAT1 += 4 * XLSTR;
    }
    // ---- bottom segment: l in [0,360) (lp = 360 + l) ----
    #pragma unroll 2
    for (int i = 0; i < LDIM / 4; ++i) {
        v2f a0, a1, b;
        b.x  = pBB[0];       b.y  = pBB[NLATC];
        a0.x = pAB0[0];      a0.y = pAB0[XLSTR];
        a1.x = pAB1[0];      a1.y = pAB1[XLSTR];
        accB0 = wmma4(a0, b, accB0);
        accB1 = wmma4(a1, b, accB1);
        pBB  += 4 * NLATC;
        pAB0 += 4 * XLSTR;  pAB1 += 4 * XLSTR;
    }

    // ---- epilogue: apply per-lane column signs, route to Re/Im planes ----
    if (valid) {
        const int st     = right;                 // 0 = s, 1 = t
        const int plane  = right ^ g;             // 0 = Re, 1 = Im
        const int kprime = plane * MPAD + m;
        float* __restrict__ Xp =
            X + ((size_t)(kprime >> 1) * ROWS2) * 2 + (kprime & 1);
        #pragma unroll
        for (int v = 0; v < 8; ++v) {
            const int bc0 = v + 8 * half;         // tile 0 rows 0..15
            const size_t r2a = ((size_t)bc0 * 2 + st) * NLATC + kc;
            const size_t r2b = ((size_t)(bc0 + 16) * 2 + st) * NLATC + kc;
            Xp[r2a * 2] = fmaf(sT, accT0[v], sB * accB0[v]);
            Xp[r2b * 2] = fmaf(sT, accT1[v], sB * accB1[v]);
        }
    }
}

// ---------------- DFT weight table (irfft as GEMM), pair-interleaved ----------------
__global__ void __launch_bounds__(256)
sht_fill_w(float* __restrict__ W /* [XK/2][NLON][2] */) {
    const int idx = blockIdx.x * blockDim.x + threadIdx.x;
    if (idx >= XK * NLON_C) return;
    const int k1 = idx & 1;
    const int u  = idx >> 1;
    const int kh = u / NLON_C;
    const int j  = u - kh * NLON_C;
    const int kprime = kh * 2 + k1;
    const int plane  = (kprime >= MPAD) ? 1 : 0;  // 0 = Re weights, 1 = Im weights
    const int m      = kprime - plane * MPAD;
    float val = 0.f;                              // K padding rows stay zero
    if (m <= 360) {
        if (m == 0)        val = plane ? 0.f : 1.f;                     // DC (imag ignored)
        else if (m == 360) val = plane ? 0.f : ((j & 1) ? -1.f : 1.f);  // Nyquist
        else {
            const int   mj  = (m * j) % NLON_C;                         // exact reduction
            const float ang = (float)mj * (6.28318530717958647692f / 720.f);
            val = plane ? (-2.f * sinf(ang)) : (2.f * cosf(ang));
        }
    }
    W[idx] = val;
}

// ---------------- stage 2: irfft GEMM (46208 x 728) x (728 x 720) ----------------
// grid = (45, 181), block = 256 (8 waves); wave -> one 32x16 tile, b64 fragments
__global__ void __launch_bounds__(256)
sht_stage2(const float* __restrict__ X, const float* __restrict__ W,
           float* __restrict__ out) {
    const int lane = threadIdx.x & 31;
    const int wave = threadIdx.x >> 5;
    const int half = lane >> 4;
    const int r    = lane & 15;
    const int tm   = blockIdx.y * 8 + wave;       // 32-row tiles
    if (tm >= TM2) return;                        // wave-uniform exit
    const int tn      = blockIdx.x;               // 0..44
    const int rowbase = tm * 32;
    const int n0      = tn * 16;

    // pair index kh = k0/2 + half; element pair is one aligned b64 per lane
    const float* Ap = X + ((size_t)half * ROWS2 + rowbase + r) * 2;
    const float* Bp = W + ((size_t)half * NLON_C + n0 + r) * 2;

    v8f acc0 = {}, acc1 = {};
    #pragma unroll 2
    for (int k0 = 0; k0 < XK; k0 += 4) {
        v2f b  = *(const v2f*)Bp;
        v2f a0 = *(const v2f*)Ap;
        v2f a1 = *(const v2f*)(Ap + 32);          // rows +16
        acc0 = wmma4(a0, b, acc0);
        acc1 = wmma4(a1, b, acc1);
        Ap += (size_t)2 * ROWS2 * 2;
        Bp += (size_t)2 * NLON_C * 2;
    }

    #pragma unroll
    for (int v = 0; v < 8; ++v) {
        const size_t row0 = (size_t)(rowbase + v + 8 * half);
        out[row0 * NLON_C + n0 + r]        = acc0[v];   // coalesced
        out[(row0 + 16) * NLON_C + n0 + r] = acc1[v];
    }
}

// ---------------- launcher ----------------
extern "C" void kernel_launch(void* const* d_in, const int* in_sizes, int n_in,
                              void* d_out, int out_size, void* d_ws, size_t ws_size,
                              hipStream_t stream) {
    (void)in_sizes; (void)n_in; (void)out_size; (void)ws_size;
    const float* x    = (const float*)d_in[0];   // (4,8,2,360,361,2) f32
    const float* dpct = (const float*)d_in[1];   // (2,361,360,361)   f32
    float* out = (float*)d_out;                  // (4,8,2,361,720)   f32

    float* X = (float*)d_ws;                            // XK * ROWS2 floats
    float* W = X + (size_t)XK * ROWS2;                  // XK * NLON  floats

    const int wtot = XK * NLON_C;                       // 524,160
    sht_fill_w<<<(wtot + 255) / 256, 256, 0, stream>>>(W);
    sht_stage1<<<dim3(6, MM, 2), 256, 0, stream>>>(x, dpct, X);
    sht_stage2<<<dim3(45, 181), 256, 0, stream>>>(X, W, out);
}